// StructuredDeepLinear_7584912245164
// MI455X (gfx1250) — compile-verified
//
#include <hip/hip_runtime.h>

#define DIM 1024
#define MB 32
#define NDEPTH 1000
#define NSTAGE (2 * NDEPTH)
#define ROWS 16
#define RS 1028   // padded LDS row stride in floats (1028 % 64 == 4 -> conflict-free frag loads)

typedef __attribute__((ext_vector_type(2))) float v2f;
typedef __attribute__((ext_vector_type(8))) float v8f;

// Split workgroup barrier: make our LDS stores visible, signal arrival.
// Independent global loads are issued between signal and wait so their latency
// overlaps the barrier convergence of all 32 waves.
__device__ __forceinline__ void wg_barrier_signal_lds() {
    asm volatile("s_wait_dscnt 0x0\n\t"
                 "s_barrier_signal -1" ::: "memory");
}
__device__ __forceinline__ void wg_barrier_wait() {
    asm volatile("s_barrier_wait -1" ::: "memory");
}

__global__ __launch_bounds__(1024, 1)
void monarch_chain(const float* __restrict__ X,
                   const float* __restrict__ L,
                   const float* __restrict__ R,
                   float* __restrict__ Out)
{
    // two ping-pong activation buffers: 2 * 16 rows * 1028 floats = 128.5 KB LDS
    __shared__ float buf[2 * ROWS * RS];

    const int tid  = threadIdx.x;
    const int lane = tid & 31;
    const int wave = tid >> 5;            // Monarch block index n = 0..31
    const int l15  = lane & 15;
    const int half = lane >> 4;           // 0 / 1 (upper half-wave handles K+2/K+3)
    const int moff = half * 2;            // K offset inside a 4-wide k-step
    const int row0 = blockIdx.x * ROWS;   // this WG's batch tile

    // ---- load activation tile x[row0:row0+16, :] into buf[0] (padded rows) ----
    {
        const int r  = tid >> 6;          // 0..15
        const int c0 = (tid & 63) * 16;   // 16 floats per thread, float4-vectorized
        const float4* src = (const float4*)(X + (size_t)(row0 + r) * DIM + c0);
        float4* dst = (float4*)(&buf[r * RS + c0]);
        dst[0] = src[0]; dst[1] = src[1]; dst[2] = src[2]; dst[3] = src[3];
    }

    // weight block base for stage t: L/R [d][n=wave][k][m], row-major 32x32
    auto wptr = [&](int t) -> const float* {
        const float* base = (t & 1) ? R : L;
        return base + ((size_t)((t >> 1) * MB + wave)) * (MB * MB);
    };
    // per-lane A-fragment (weight) offsets:  A[c_local][m] with c rows of stride 32
    const int wlane0 = l15 * MB + moff;          // column half c0 = 0
    const int wlane1 = (16 + l15) * MB + moff;   // column half c0 = 16

    // double-buffered weight fragments (16 x v2f each): software pipeline across stages
    v2f wA[16], wB[16];
    {
        const float* w0 = wptr(0);
        #pragma unroll
        for (int j = 0; j < 8; ++j) {
            wA[j]     = *(const v2f*)(w0 + wlane0 + 4 * j);
            wA[8 + j] = *(const v2f*)(w0 + wlane1 + 4 * j);
        }
    }

    __syncthreads();   // x tile visible to all waves

    const int bload  = l15 * RS + wave * MB + moff;  // B-fragment (activation) LDS offset
    const int dstore = l15 * RS + wave;              // transposed D-store LDS base

    auto stage = [&](int t, v2f* wcur, v2f* wnxt) {
        const int sread  = (t & 1) * (ROWS * RS);
        const int swrite = ((t + 1) & 1) * (ROWS * RS);

        // ---- activation B fragments from LDS: B[m][b] = h[b][m0+m], conflict-free b64
        v2f act[8];
        {
            const float* s = &buf[sread + bload];
            #pragma unroll
            for (int j = 0; j < 8; ++j)
                act[j] = *(const v2f*)(s + 4 * j);
        }

        // ---- D[c][b] = sum_m W[c][m] * h[b][m]  via 16x V_WMMA_F32_16X16X4_F32 ----
        v8f acc0 = {0.f, 0.f, 0.f, 0.f, 0.f, 0.f, 0.f, 0.f};
        v8f acc1 = acc0;
        #pragma unroll
        for (int j = 0; j < 8; ++j) {
            acc0 = __builtin_amdgcn_wmma_f32_16x16x4_f32(
                false, wcur[j],     false, act[j], (short)0, acc0, false, false);
            acc1 = __builtin_amdgcn_wmma_f32_16x16x4_f32(
                false, wcur[8 + j], false, act[j], (short)0, acc1, false, false);
        }

        // ---- Monarch permutation fused into the store: dst[b][c*32 + n] = D[c][b] ----
        {
            float* d = &buf[swrite + dstore];
            #pragma unroll
            for (int v = 0; v < 8; ++v) {
                const int c = v + half * 8;      // D layout: VGPR v -> M = v (+8 upper half)
                d[c * MB]        = acc0[v];      // columns 0..15
                d[(c + 16) * MB] = acc1[v];      // columns 16..31
            }
        }

        // ---- split barrier: signal, then issue next-stage weight loads so their
        //      L2 latency overlaps the barrier convergence of all 32 waves ----
        wg_barrier_signal_lds();
        {
            const int tn = (t + 1 < NSTAGE) ? (t + 1) : t;
            const float* wn = wptr(tn);
            #pragma unroll
            for (int j = 0; j < 8; ++j) {
                wnxt[j]     = *(const v2f*)(wn + wlane0 + 4 * j);
                wnxt[8 + j] = *(const v2f*)(wn + wlane1 + 4 * j);
            }
            // prefetch two layers ahead into L2 (32 lanes * 128B = this wave's 4KB block)
            const int tp = (t + 4 < NSTAGE) ? (t + 4) : t;
            __builtin_prefetch((const void*)(wptr(tp) + lane * MB), 0, 1);
        }
        wg_barrier_wait();
    };

    for (int t = 0; t < NSTAGE; t += 2) {
        stage(t,     wA, wB);
        stage(t + 1, wB, wA);
    }

    // ---- after an even number of stages the result sits in buf[0]; write out ----
    {
        const int r  = tid >> 6;
        const int c0 = (tid & 63) * 16;
        const float4* src = (const float4*)(&buf[r * RS + c0]);
        float4* dst = (float4*)(Out + (size_t)(row0 + r) * DIM + c0);
        dst[0] = src[0]; dst[1] = src[1]; dst[2] = src[2]; dst[3] = src[3];
    }
}

extern "C" void kernel_launch(void* const* d_in, const int* in_sizes, int n_in,
                              void* d_out, int out_size, void* d_ws, size_t ws_size,
                              hipStream_t stream)
{
    const float* X = (const float*)d_in[0];
    const float* L = (const float*)d_in[1];
    const float* R = (const float*)d_in[2];
    float* Out = (float*)d_out;

    dim3 grid(512 / ROWS);   // 32 workgroups: one 16-row batch tile each (independent chains)
    dim3 block(1024);        // 32 waves: one wave per Monarch block n
    monarch_chain<<<grid, block, 0, stream>>>(X, L, R, Out);

    (void)in_sizes; (void)n_in; (void)out_size; (void)d_ws; (void)ws_size;
}